// AtlasTGN_31911607009495
// MI455X (gfx1250) — compile-verified
//
// AtlasTGN for MI455X (gfx1250) — fp32 WMMA (V_WMMA_F32_16X16X4_F32) fused TGN pipeline.
//
// Roofline: ~35 GFLOP fp32 vs ~800 MB HBM traffic (incl. the mandatory 256MB
// memory copy into d_out) -> bandwidth-bound (~34us floor at 23.3 TB/s).
// fp32 WMMA keeps reference numerics while putting all GEMMs on the matrix
// pipes; activations/gathers ride the vector/memory pipes.
//
// v2: removed the divergent `if (row < TM)` guards around A-operand LDS loads
// in the Q/output GEMMs (they emitted per-K-step saveexec/branch/exec-restore).
// Rows 8..15 of the 16-row WMMA tile are discarded at the epilogue, so the
// LDS read row is simply clamped (row & 7) — straight-line ds_load_b64 now.

#include <hip/hip_runtime.h>
#include <math.h>

#define D_   128
#define DE_  128
#define DT_  100
#define F_   10
#define B_   512
#define G_   1536
#define N1_  15360
#define NN_  500000

#define TM   8            // dst rows per workgroup in fused attention
#define KVR  (TM * F_)    // 80 kv rows per workgroup
#define QK   (D_ + DT_)   // 228
#define KDIM (D_ + DE_ + DT_) // 356
#define OK_  (2 * D_)     // 256

typedef float v2f __attribute__((ext_vector_type(2)));
typedef float v8f __attribute__((ext_vector_type(8)));

// D = A(16x4,f32) * B(4x16,f32) + C(16x16,f32), wave32.
// Lane layout (ISA 7.12.2): A: lane<16 -> row=lane, K={k0,k0+1}; lane>=16 ->
// row=lane-16, K={k0+2,k0+3}. B mirrored: col=lane&15, K rows likewise.
// C/D: element j of lane l -> (M = j + 8*(l>=16), N = l&15).
__device__ __forceinline__ v8f wmma4(v2f a, v2f b, v8f c) {
  return __builtin_amdgcn_wmma_f32_16x16x4_f32(false, a, false, b, (short)0, c,
                                               false, false);
}

__device__ __forceinline__ v2f ld2(const float* p) {
  const float2 t = *(const float2*)p;
  v2f r; r.x = t.x; r.y = t.y; return r;
}

// ---------------------------------------------------------------------------
// Fused TGN attention layer. One workgroup = TM(8) dst rows, 256 threads.
// out[n,:] = relu([dst_h | softmax-attn(o)] @ Wo + bo)
// ---------------------------------------------------------------------------
__global__ __launch_bounds__(256) void attn_fused(
    const float* __restrict__ dst_feat,   // [N,128]
    const float* __restrict__ src_feat,   // [N,F,128]
    const int*   __restrict__ dst_nodes,  // [N]
    const int*   __restrict__ src_nodes,  // [N,F]
    const float* __restrict__ dst_ts,     // [N]
    const float* __restrict__ src_ts,     // [N,F]
    const float* __restrict__ edge_feat,  // [N,F,128]
    const float* __restrict__ memory,     // [NN,128]
    const float* __restrict__ Wq, const float* __restrict__ bq,   // [228,128]
    const float* __restrict__ Wk, const float* __restrict__ bk,   // [356,128]
    const float* __restrict__ Wv, const float* __restrict__ bv,   // [356,128]
    const float* __restrict__ Wo, const float* __restrict__ bo,   // [256,128]
    const float* __restrict__ tw, const float* __restrict__ tb,   // [100]
    float* __restrict__ out)              // [N,128]
{
  __shared__ float s_dsth[TM * 132];      // dst_h, padded stride (132%64==4)
  __shared__ float s_qo[TM * 132];        // q, later reused for o
  __shared__ float s_v[KVR * 128];        // all v rows
  __shared__ float s_k[16 * 132];         // one k chunk
  __shared__ float s_sc[KVR * 2];         // scores [kvrow][head]
  __shared__ float s_aw[TM * 2 * F_];     // softmax weights
  __shared__ float s_dt[KVR];
  __shared__ int   s_sn[KVR];
  __shared__ float s_qt[DT_];             // q_time = cos(tb)

  const int tid  = threadIdx.x;
  const int lane = tid & 31;
  const int wv   = tid >> 5;
  const int n0   = blockIdx.x * TM;

  // ---- preload: neighbor ids, dt, q_time, dst_h (feat + memory gather) ----
  if (tid < KVR) {
    const int i = tid / F_, f = tid - i * F_;
    const int nn = n0 + i;
    s_sn[tid] = src_nodes[nn * F_ + f];
    s_dt[tid] = dst_ts[nn] - src_ts[nn * F_ + f];
  }
  if (tid < DT_) s_qt[tid] = cosf(tb[tid]);
  for (int idx = tid; idx < TM * D_; idx += 256) {
    const int r = idx >> 7, c = idx & 127;
    const int dn = dst_nodes[n0 + r];
    s_dsth[r * 132 + c] =
        dst_feat[(size_t)(n0 + r) * D_ + c] + memory[(size_t)dn * D_ + c];
  }
  __syncthreads();

  const int row  = lane & 15;
  const int rowc = lane & (TM - 1);   // clamped row for A-LDS reads; WMMA rows
                                      // 8..15 are duplicates, discarded later
  const int koff = (lane >> 4) * 2;
  const int col  = wv * 16 + (lane & 15);

  // ---- Q projection: q = [dst_h | cos(tb)] @ Wq + bq  (M=16, N=128, K=228) -
  {
    v8f acc = {0.f, 0.f, 0.f, 0.f, 0.f, 0.f, 0.f, 0.f};
    for (int k0 = 0; k0 < D_; k0 += 4) {
      const int kk = k0 + koff;
      const v2f a = ld2(&s_dsth[rowc * 132 + kk]);
      v2f b; b.x = Wq[(size_t)kk * D_ + col];
             b.y = Wq[(size_t)(kk + 1) * D_ + col];
      acc = wmma4(a, b, acc);
    }
    for (int k0 = D_; k0 < QK; k0 += 4) {
      const int kk = k0 + koff;
      v2f a; a.x = s_qt[kk - D_]; a.y = s_qt[kk - D_ + 1];  // row-independent
      v2f b; b.x = Wq[(size_t)kk * D_ + col];
             b.y = Wq[(size_t)(kk + 1) * D_ + col];
      acc = wmma4(a, b, acc);
    }
    if (lane < 16) {                       // valid D rows are M=0..7 (j<8)
      const float bb = bq[col];
      #pragma unroll
      for (int j = 0; j < TM; ++j) s_qo[j * 132 + col] = acc[j] + bb;
    }
  }
  __syncthreads();

  // ---- K/V projections in 16-row chunks; scores per chunk -----------------
  for (int c = 0; c < KVR / 16; ++c) {
    {
      const int gr = c * 16 + row;         // kv row in workgroup [0,80)
      const int i  = gr / F_, f = gr - i * F_;
      const int nn = n0 + i;
      const float dt = s_dt[gr];
      const int   sn = s_sn[gr];
      const float* psf = src_feat + (size_t)(nn * F_ + f) * D_;
      const float* pmm = memory + (size_t)sn * D_;
      const float* pef = edge_feat + (size_t)(nn * F_ + f) * DE_;

      v8f ak = {0.f, 0.f, 0.f, 0.f, 0.f, 0.f, 0.f, 0.f};
      v8f av = {0.f, 0.f, 0.f, 0.f, 0.f, 0.f, 0.f, 0.f};

      // region 1: src_h = src_feat + memory[src_node]
      for (int k0 = 0; k0 < D_; k0 += 4) {
        const int kk = k0 + koff;
        const v2f t1 = ld2(psf + kk), t2 = ld2(pmm + kk);
        v2f a; a.x = t1.x + t2.x; a.y = t1.y + t2.y;
        v2f b1, b2;
        b1.x = Wk[(size_t)kk * D_ + col]; b1.y = Wk[(size_t)(kk + 1) * D_ + col];
        b2.x = Wv[(size_t)kk * D_ + col]; b2.y = Wv[(size_t)(kk + 1) * D_ + col];
        ak = wmma4(a, b1, ak);
        av = wmma4(a, b2, av);
      }
      // region 2: edge features
      for (int k0 = D_; k0 < D_ + DE_; k0 += 4) {
        const int kk = k0 + koff;
        const v2f a = ld2(pef + (kk - D_));
        v2f b1, b2;
        b1.x = Wk[(size_t)kk * D_ + col]; b1.y = Wk[(size_t)(kk + 1) * D_ + col];
        b2.x = Wv[(size_t)kk * D_ + col]; b2.y = Wv[(size_t)(kk + 1) * D_ + col];
        ak = wmma4(a, b1, ak);
        av = wmma4(a, b2, av);
      }
      // region 3: time encoding cos(dt*tw + tb)
      for (int k0 = D_ + DE_; k0 < KDIM; k0 += 4) {
        const int kk = k0 + koff;
        const int t = kk - (D_ + DE_);
        v2f a;
        a.x = cosf(fmaf(dt, tw[t], tb[t]));
        a.y = cosf(fmaf(dt, tw[t + 1], tb[t + 1]));
        v2f b1, b2;
        b1.x = Wk[(size_t)kk * D_ + col]; b1.y = Wk[(size_t)(kk + 1) * D_ + col];
        b2.x = Wv[(size_t)kk * D_ + col]; b2.y = Wv[(size_t)(kk + 1) * D_ + col];
        ak = wmma4(a, b1, ak);
        av = wmma4(a, b2, av);
      }
      const float bbk = bk[col], bbv = bv[col];
      const int mbase = (lane >> 4) * 8;
      #pragma unroll
      for (int j = 0; j < 8; ++j) {
        const int M = mbase + j;
        s_k[M * 132 + col] = ak[j] + bbk;
        s_v[(c * 16 + M) * 128 + col] = av[j] + bbv;
      }
    }
    __syncthreads();
    // scores for these 16 kv rows: q . k per head, * DH^-0.5
    if (tid < 32) {
      const int r = tid & 15, h = tid >> 4;
      const int gr = c * 16 + r;
      const int i = gr / F_;
      const float* qp = &s_qo[i * 132 + h * 64];
      const float* kp = &s_k[r * 132 + h * 64];
      float s = 0.f;
      #pragma unroll 8
      for (int d2 = 0; d2 < 64; ++d2) s = fmaf(qp[d2], kp[d2], s);
      s_sc[gr * 2 + h] = s * 0.125f;       // 64^-0.5
    }
    __syncthreads();
  }

  // ---- softmax over F per (row, head) -------------------------------------
  if (tid < TM * 2) {
    const int r = tid >> 1, h = tid & 1;
    float m = -1e30f;
    for (int f = 0; f < F_; ++f) m = fmaxf(m, s_sc[(r * F_ + f) * 2 + h]);
    float e[F_], sum = 0.f;
    for (int f = 0; f < F_; ++f) { e[f] = expf(s_sc[(r * F_ + f) * 2 + h] - m); sum += e[f]; }
    const float inv = 1.f / sum;
    for (int f = 0; f < F_; ++f) s_aw[(r * 2 + h) * F_ + f] = e[f] * inv;
  }
  __syncthreads();

  // ---- o = sum_f a * v  (q is dead; reuse s_qo) ---------------------------
  for (int idx = tid; idx < TM * D_; idx += 256) {
    const int r = idx >> 7, d = idx & 127, h = d >> 6;
    const float* ap = &s_aw[(r * 2 + h) * F_];
    float s = 0.f;
    #pragma unroll
    for (int f = 0; f < F_; ++f) s = fmaf(ap[f], s_v[(r * F_ + f) * 128 + d], s);
    s_qo[r * 132 + d] = s;
  }
  __syncthreads();

  // ---- out = relu([dst_h | o] @ Wo + bo)  (M=16, N=128, K=256) ------------
  {
    v8f acc = {0.f, 0.f, 0.f, 0.f, 0.f, 0.f, 0.f, 0.f};
    for (int k0 = 0; k0 < D_; k0 += 4) {
      const int kk = k0 + koff;
      const v2f a = ld2(&s_dsth[rowc * 132 + kk]);
      v2f b; b.x = Wo[(size_t)kk * D_ + col];
             b.y = Wo[(size_t)(kk + 1) * D_ + col];
      acc = wmma4(a, b, acc);
    }
    for (int k0 = D_; k0 < OK_; k0 += 4) {
      const int kk = k0 + koff;
      const v2f a = ld2(&s_qo[rowc * 132 + (kk - D_)]);
      v2f b; b.x = Wo[(size_t)kk * D_ + col];
             b.y = Wo[(size_t)(kk + 1) * D_ + col];
      acc = wmma4(a, b, acc);
    }
    if (lane < 16) {
      const float bb = bo[col];
      #pragma unroll
      for (int j = 0; j < TM; ++j) {
        const float v = acc[j] + bb;
        out[(size_t)(n0 + j) * D_ + col] = v > 0.f ? v : 0.f;
      }
    }
  }
}

// ---------------------------------------------------------------------------
// Generic fp32-WMMA GEMM: C[M,N] = act(A[M,K] @ W[K,N] + bias)
// grid = (M/16, N/128), block = 256 (8 waves, one 16-col tile each)
// ---------------------------------------------------------------------------
__global__ __launch_bounds__(256) void gemm_wmma(
    const float* __restrict__ A, int K,
    const float* __restrict__ W, int N,
    const float* __restrict__ bias,
    float* __restrict__ C, int relu)
{
  const int lane = threadIdx.x & 31;
  const int wv   = threadIdx.x >> 5;
  const int m0   = blockIdx.x * 16;
  const int col  = blockIdx.y * 128 + wv * 16 + (lane & 15);
  const int row  = lane & 15;
  const int koff = (lane >> 4) * 2;

  const float* Ap = A + (size_t)(m0 + row) * K + koff;
  v8f acc = {0.f, 0.f, 0.f, 0.f, 0.f, 0.f, 0.f, 0.f};
  for (int k0 = 0; k0 < K; k0 += 4) {
    const v2f a = ld2(Ap + k0);
    v2f b; b.x = W[(size_t)(k0 + koff) * N + col];
           b.y = W[(size_t)(k0 + koff + 1) * N + col];
    acc = wmma4(a, b, acc);
  }
  const float bb = bias ? bias[col] : 0.f;
  const int mbase = (lane >> 4) * 8;
  #pragma unroll
  for (int j = 0; j < 8; ++j) {
    float v = acc[j] + bb;
    if (relu) v = v > 0.f ? v : 0.f;
    C[(size_t)(m0 + mbase + j) * N + col] = v;
  }
}

// h_root1[g,d] = mean_f h1[g*F+f, d]
__global__ void hroot_mean(const float* __restrict__ h1, float* __restrict__ hr) {
  const int idx = blockIdx.x * 256 + threadIdx.x;
  if (idx >= G_ * D_) return;
  const int g = idx >> 7, d = idx & 127;
  float s = 0.f;
  #pragma unroll
  for (int f = 0; f < F_; ++f) s += h1[(size_t)(g * F_ + f) * D_ + d];
  hr[idx] = s * 0.1f;
}

// h_total = (h0 + h_root1) / 2, in place over h0
__global__ void combine_half(float* __restrict__ h0, const float* __restrict__ hr) {
  const int idx = blockIdx.x * 256 + threadIdx.x;
  if (idx >= G_ * D_) return;
  h0[idx] = 0.5f * (h0[idx] + hr[idx]);
}

// Build GRU input x = [msg | prev_mem | cos(dt*mt_w+mt_b) | edge_feat], and pmem
__global__ void build_gru_x(
    const float* __restrict__ h_total, const int* __restrict__ nodes,
    const float* __restrict__ root_ts, const float* __restrict__ memory_ts,
    const float* __restrict__ memory, const float* __restrict__ root_ef,
    const float* __restrict__ mt_w, const float* __restrict__ mt_b,
    float* __restrict__ xbuf, float* __restrict__ pmem)
{
  const int i = blockIdx.x;        // [0,1024)
  const int t = threadIdx.x;       // [0,128)
  const int node = nodes[i];
  const float dtm = fmaxf(root_ts[i & (B_ - 1)] - memory_ts[node], 0.f);
  float* xr = xbuf + (size_t)i * 484;
  xr[t] = h_total[(size_t)i * D_ + t];
  const float pm = memory[(size_t)node * D_ + t];
  xr[128 + t] = pm;
  pmem[(size_t)i * D_ + t] = pm;
  if (t < DT_) xr[256 + t] = cosf(fmaf(dtm, mt_w[t], mt_b[t]));
  xr[356 + t] = root_ef[(size_t)(i & (B_ - 1)) * DE_ + t];
}

// GRU gate combine + deterministic scatter (last duplicate index wins)
__global__ void gru_combine(
    const float* __restrict__ gi, const float* __restrict__ gh,
    const float* __restrict__ pmem, const int* __restrict__ nodes,
    const float* __restrict__ root_ts,
    float* __restrict__ out_mem, float* __restrict__ out_ts)
{
  __shared__ int skip;
  const int i = blockIdx.x;        // [0,1024)
  const int d = threadIdx.x;       // [0,128)
  const int node = nodes[i];
  if (d == 0) {
    int sk = 0;
    for (int j = i + 1; j < 2 * B_; ++j)
      if (nodes[j] == node) { sk = 1; break; }
    skip = sk;
  }
  __syncthreads();
  if (skip) return;
  const size_t gb = (size_t)i * 384;
  const float ir = gi[gb + d],        hr = gh[gb + d];
  const float iz = gi[gb + 128 + d],  hz = gh[gb + 128 + d];
  const float in_ = gi[gb + 256 + d], hn = gh[gb + 256 + d];
  const float r = 1.f / (1.f + expf(-(ir + hr)));
  const float z = 1.f / (1.f + expf(-(iz + hz)));
  const float n = tanhf(fmaf(r, hn, in_));
  const float h = pmem[(size_t)i * D_ + d];
  out_mem[(size_t)node * D_ + d] = (1.f - z) * n + z * h;
  if (d == 0) out_ts[node] = root_ts[i & (B_ - 1)];
}

// pos/neg = relu(s + t) @ Wout + bout
__global__ void ep_final(
    const float* __restrict__ s, const float* __restrict__ td,
    const float* __restrict__ tn, const float* __restrict__ Wout,
    const float* __restrict__ bout, float* __restrict__ out)
{
  const int i = blockIdx.x * 64 + threadIdx.x;
  if (i >= B_) return;
  float p = 0.f, q = 0.f;
  for (int d = 0; d < D_; ++d) {
    const float sv = s[(size_t)i * D_ + d];
    const float w = Wout[d];
    float a = sv + td[(size_t)i * D_ + d]; a = a > 0.f ? a : 0.f;
    float b = sv + tn[(size_t)i * D_ + d]; b = b > 0.f ? b : 0.f;
    p = fmaf(a, w, p);
    q = fmaf(b, w, q);
  }
  out[i] = p + bout[0];
  out[B_ + i] = q + bout[0];
}

extern "C" void kernel_launch(void* const* d_in, const int* in_sizes, int n_in,
                              void* d_out, int out_size, void* d_ws, size_t ws_size,
                              hipStream_t stream) {
  const int*   dst_nodes0 = (const int*)d_in[0];
  const int*   src_nodes0 = (const int*)d_in[1];
  const float* dst_feat0  = (const float*)d_in[2];
  const float* dst_ts0    = (const float*)d_in[3];
  const float* src_ts0    = (const float*)d_in[4];
  const float* edge_feat0 = (const float*)d_in[5];
  const int*   dst_nodes1 = (const int*)d_in[6];
  const int*   src_nodes1 = (const int*)d_in[7];
  const float* dst_feat1  = (const float*)d_in[8];
  const float* src_feat1  = (const float*)d_in[9];
  const float* dst_ts1    = (const float*)d_in[10];
  const float* src_ts1    = (const float*)d_in[11];
  const float* edge_feat1 = (const float*)d_in[12];
  const float* root_ts    = (const float*)d_in[13];
  const float* root_ef    = (const float*)d_in[14];
  const float* memory     = (const float*)d_in[15];
  const float* memory_ts  = (const float*)d_in[16];
  const float* attn_Wq    = (const float*)d_in[17];
  const float* attn_bq    = (const float*)d_in[18];
  const float* attn_Wk    = (const float*)d_in[19];
  const float* attn_bk    = (const float*)d_in[20];
  const float* attn_Wv    = (const float*)d_in[21];
  const float* attn_bv    = (const float*)d_in[22];
  const float* attn_Wo    = (const float*)d_in[23];
  const float* attn_bo    = (const float*)d_in[24];
  const float* attn_tw    = (const float*)d_in[25];
  const float* attn_tb    = (const float*)d_in[26];
  const float* e2n_W      = (const float*)d_in[27];
  const float* e2n_b      = (const float*)d_in[28];
  const float* mt_w       = (const float*)d_in[29];
  const float* mt_b       = (const float*)d_in[30];
  const float* gru_Wih    = (const float*)d_in[31];
  const float* gru_Whh    = (const float*)d_in[32];
  const float* gru_bih    = (const float*)d_in[33];
  const float* gru_bhh    = (const float*)d_in[34];
  const float* ep_Wsrc    = (const float*)d_in[35];
  const float* ep_bsrc    = (const float*)d_in[36];
  const float* ep_Wdst    = (const float*)d_in[37];
  const float* ep_bdst    = (const float*)d_in[38];
  const float* ep_Wout    = (const float*)d_in[39];
  const float* ep_bout    = (const float*)d_in[40];

  // workspace layout (floats)
  float* ws = (float*)d_ws;
  size_t o = 0;
  float* h1       = ws + o; o += (size_t)N1_ * D_;
  float* next_src = ws + o; o += (size_t)N1_ * D_;
  float* hroot    = ws + o; o += (size_t)G_ * D_;
  float* h0       = ws + o; o += (size_t)G_ * D_;   // becomes h_total in place
  float* xbuf     = ws + o; o += (size_t)2 * B_ * 484;
  float* pmem     = ws + o; o += (size_t)2 * B_ * D_;
  float* gi       = ws + o; o += (size_t)2 * B_ * 384;
  float* gh       = ws + o; o += (size_t)2 * B_ * 384;
  float* sbuf     = ws + o; o += (size_t)B_ * D_;
  float* tdbuf    = ws + o; o += (size_t)B_ * D_;
  float* tnbuf    = ws + o; o += (size_t)B_ * D_;

  float* out     = (float*)d_out;
  float* out_mem = out + 1024;                       // [NN,128]
  float* out_ts  = out + 1024 + (size_t)NN_ * D_;    // [NN]

  // 0) base copies of memory/memory_ts into the output (scatter comes later)
  hipMemcpyAsync(out_mem, memory, (size_t)NN_ * D_ * sizeof(float),
                 hipMemcpyDeviceToDevice, stream);
  hipMemcpyAsync(out_ts, memory_ts, (size_t)NN_ * sizeof(float),
                 hipMemcpyDeviceToDevice, stream);

  // 1) level-1 attention (weights slice l=1)
  attn_fused<<<N1_ / TM, 256, 0, stream>>>(
      dst_feat1, src_feat1, dst_nodes1, src_nodes1, dst_ts1, src_ts1,
      edge_feat1, memory,
      attn_Wq + (size_t)QK * D_,  attn_bq + D_,
      attn_Wk + (size_t)KDIM * D_, attn_bk + D_,
      attn_Wv + (size_t)KDIM * D_, attn_bv + D_,
      attn_Wo + (size_t)OK_ * D_, attn_bo + D_,
      attn_tw + DT_, attn_tb + DT_, h1);

  // 2) h_root1 = mean over F
  hroot_mean<<<(G_ * D_ + 255) / 256, 256, 0, stream>>>(h1, hroot);

  // 3) next_src = h1 @ e2n_W + e2n_b
  gemm_wmma<<<dim3(N1_ / 16, 1), 256, 0, stream>>>(h1, D_, e2n_W, D_, e2n_b,
                                                   next_src, 0);

  // 4) level-0 attention (weights slice l=0), src features = next_src
  attn_fused<<<G_ / TM, 256, 0, stream>>>(
      dst_feat0, next_src, dst_nodes0, src_nodes0, dst_ts0, src_ts0,
      edge_feat0, memory,
      attn_Wq, attn_bq, attn_Wk, attn_bk, attn_Wv, attn_bv, attn_Wo, attn_bo,
      attn_tw, attn_tb, h0);

  // 5) h_total = (h0 + h_root1)/2
  combine_half<<<(G_ * D_ + 255) / 256, 256, 0, stream>>>(h0, hroot);

  // 6) GRU memory update for the first 2B rows
  build_gru_x<<<2 * B_, 128, 0, stream>>>(h0, dst_nodes0, root_ts, memory_ts,
                                          memory, root_ef, mt_w, mt_b, xbuf, pmem);
  gemm_wmma<<<dim3(2 * B_ / 16, 3), 256, 0, stream>>>(xbuf, 484, gru_Wih, 384,
                                                      gru_bih, gi, 0);
  gemm_wmma<<<dim3(2 * B_ / 16, 3), 256, 0, stream>>>(pmem, D_, gru_Whh, 384,
                                                      gru_bhh, gh, 0);
  gru_combine<<<2 * B_, 128, 0, stream>>>(gi, gh, pmem, dst_nodes0, root_ts,
                                          out_mem, out_ts);

  // 7) edge predictor
  gemm_wmma<<<dim3(B_ / 16, 1), 256, 0, stream>>>(h0, D_, ep_Wsrc, D_, ep_bsrc,
                                                  sbuf, 0);
  gemm_wmma<<<dim3(B_ / 16, 1), 256, 0, stream>>>(h0 + (size_t)B_ * D_, D_,
                                                  ep_Wdst, D_, ep_bdst, tdbuf, 0);
  gemm_wmma<<<dim3(B_ / 16, 1), 256, 0, stream>>>(h0 + (size_t)2 * B_ * D_, D_,
                                                  ep_Wdst, D_, ep_bdst, tnbuf, 0);
  ep_final<<<(B_ + 63) / 64, 64, 0, stream>>>(sbuf, tdbuf, tnbuf, ep_Wout,
                                              ep_bout, out);
}